// EdgeNet_3848290697764
// MI455X (gfx1250) — compile-verified
//
#include <hip/hip_runtime.h>
#include <math.h>

typedef __attribute__((ext_vector_type(2))) float v2f;
typedef __attribute__((ext_vector_type(8))) float v8f;

#define N_ROWS 16384
#define DIM 20

// Kernel 1: h = x @ W^T + b  (torch nn.Linear convention), sq[n] = ||h_n||^2
__global__ void __launch_bounds__(256)
linear_rownorm_kernel(const float* __restrict__ x, const float* __restrict__ W,
                      const float* __restrict__ b, float* __restrict__ h,
                      float* __restrict__ sq) {
  int n = blockIdx.x * blockDim.x + threadIdx.x;
  if (n >= N_ROWS) return;
  float xr[DIM];
#pragma unroll
  for (int k = 0; k < DIM; ++k) xr[k] = x[n * DIM + k];
  float s = 0.0f;
#pragma unroll
  for (int d = 0; d < DIM; ++d) {
    float acc = b[d];
#pragma unroll
    for (int k = 0; k < DIM; ++k) acc = fmaf(xr[k], W[d * DIM + k], acc);
    h[n * DIM + d] = acc;
    s = fmaf(acc, acc, s);
  }
  sq[n] = s;
}

// Kernel 2: dist[i,j] = sqrt(max(sq[i] + sq[j] - 2 * (h @ h^T)[i,j], 0))
// Block = 256 threads = 8 waves; block tile = 128x128 of the output.
// Wave w owns rows [m0, m0+16) x cols [n0, n0+128) as 8 WMMA 16x16 tiles.
// K = 20 = 5 steps of V_WMMA_F32_16X16X4_F32.
__global__ void __launch_bounds__(256)
gram_dist_wmma_kernel(const float* __restrict__ h, const float* __restrict__ sq,
                      float* __restrict__ out) {
  const int lane = threadIdx.x & 31;
  const int wave = threadIdx.x >> 5;   // 0..7
  const int lo = lane & 15;
  const int hi = lane >> 4;            // 0 or 1
  const int m0 = blockIdx.y * 128 + wave * 16;
  const int n0 = blockIdx.x * 128;

  // A fragment (16x4 f32 layout): lane l holds (M = m0 + (l&15), K = 4k + 2*(l>>4) + {0,1})
  const float* arow = h + (size_t)(m0 + lo) * DIM + 2 * hi;
  v2f a[5];
#pragma unroll
  for (int k = 0; k < 5; ++k) a[k] = *(const v2f*)(arow + 4 * k);

  v8f acc[8];
#pragma unroll
  for (int t = 0; t < 8; ++t) {
    // B fragment for the Gram matrix is the same slicing of h on the column tile.
    const float* brow = h + (size_t)(n0 + t * 16 + lo) * DIM + 2 * hi;
    v8f c = {0.f, 0.f, 0.f, 0.f, 0.f, 0.f, 0.f, 0.f};
#pragma unroll
    for (int k = 0; k < 5; ++k) {
      v2f bf = *(const v2f*)(brow + 4 * k);
      // (neg_a, A, neg_b, B, c_mod, C, reuse_a, reuse_b)
      c = __builtin_amdgcn_wmma_f32_16x16x4_f32(
          false, a[k], false, bf, (short)0, c, false, false);
    }
    acc[t] = c;
  }

  // C/D layout: VGPR v, lane l -> row = m0 + v + 8*(l>>4), col = n0t + (l&15)
  float sqa[8];
#pragma unroll
  for (int v = 0; v < 8; ++v) sqa[v] = sq[m0 + v + 8 * hi];

#pragma unroll
  for (int t = 0; t < 8; ++t) {
    const int col = n0 + t * 16 + lo;
    const float sqb = sq[col];
#pragma unroll
    for (int v = 0; v < 8; ++v) {
      float d2 = sqa[v] + sqb - 2.0f * acc[t][v];
      d2 = fmaxf(d2, 0.0f);                      // clamp roundoff negatives
      out[(size_t)(m0 + v + 8 * hi) * N_ROWS + col] = sqrtf(d2); // sqrt(0)=0 on diagonal
    }
  }
}

extern "C" void kernel_launch(void* const* d_in, const int* in_sizes, int n_in,
                              void* d_out, int out_size, void* d_ws, size_t ws_size,
                              hipStream_t stream) {
  const float* x = (const float*)d_in[0];  // [N, 20]
  const float* W = (const float*)d_in[1];  // [20, 20]
  const float* b = (const float*)d_in[2];  // [20]
  float* out = (float*)d_out;              // [N, N]

  float* h  = (float*)d_ws;                   // N*20 floats
  float* sq = h + (size_t)N_ROWS * DIM;       // N floats

  linear_rownorm_kernel<<<(N_ROWS + 255) / 256, 256, 0, stream>>>(x, W, b, h, sq);

  dim3 grid(N_ROWS / 128, N_ROWS / 128);
  gram_dist_wmma_kernel<<<grid, 256, 0, stream>>>(h, sq, out);
}